// SparseHyperGraphAttentionLayer_81793357185035
// MI455X (gfx1250) — compile-verified
//
#include <hip/hip_runtime.h>
#include <hip/hip_bf16.h>

#define D      128
#define AN     8
#define AE     16
#define ALPHA  0.2f
#define NEGV   (-9000000000000000.0f)

typedef float v2f __attribute__((ext_vector_type(2)));
typedef float v8f __attribute__((ext_vector_type(8)));

__device__ __forceinline__ float wave_sum(float v) {
#pragma unroll
  for (int off = 16; off > 0; off >>= 1) v += __shfl_xor(v, off, 32);
  return v;
}

__device__ __forceinline__ float lrelu(float x) { return x > 0.0f ? x : ALPHA * x; }

// ---------------------------------------------------------------------------
// C[R x 128] = A[R x 128] @ W[128 x 128]   (all row-major f32, R % 16 == 0)
// One wave32 per 16x16 output tile, K=128 via 32x V_WMMA_F32_16X16X4_F32.
// A frag (16x4): lanes 0-15 hold K=k,k+1 ; lanes 16-31 hold K=k+2,k+3.
// B frag (4x16): lane's col = c0+(lane&15); vgpr0/1 = rows k+koff, k+koff+1.
// C/D (16x16):   vgpr g -> row r0 + g + 8*(lane>=16), col c0+(lane&15).
// ---------------------------------------------------------------------------
__global__ __launch_bounds__(256) void gemm_rowmajor_128(
    const float* __restrict__ A, const float* __restrict__ W,
    float* __restrict__ C, int R) {
  const int wave = (int)((blockIdx.x * blockDim.x + threadIdx.x) >> 5);
  const int lane = (int)(threadIdx.x & 31);
  const int rowTile = wave >> 3;   // 8 column tiles per row strip (128/16)
  const int colTile = wave & 7;
  if (rowTile * 16 >= R) return;

  const int r0   = rowTile << 4;
  const int c0   = colTile << 4;
  const int mrow = r0 + (lane & 15);
  const int koff = (lane >> 4) << 1;        // 0 for lanes 0-15, 2 for 16-31
  const int col  = c0 + (lane & 15);

  const float* aRow = A + (size_t)mrow * D;
  v8f acc = {};
#pragma unroll
  for (int k = 0; k < D; k += 4) {
    v2f a = *(const v2f*)(aRow + k + koff);               // 8B-aligned
    v2f b;
    b.x = W[(size_t)(k + koff) * D + col];
    b.y = W[(size_t)(k + koff + 1) * D + col];
    acc = __builtin_amdgcn_wmma_f32_16x16x4_f32(
        /*neg_a=*/false, a, /*neg_b=*/false, b,
        /*c_mod=*/(short)0, acc, /*reuse_a=*/false, /*reuse_b=*/false);
  }

  const int rbase = r0 + ((lane >> 4) << 3);
#pragma unroll
  for (int g = 0; g < 8; ++g)
    C[(size_t)(rbase + g) * D + col] = acc[g];
}

// ---------------------------------------------------------------------------
// Edge attention: one wave32 per edge. Lane j owns cols [4j, 4j+4).
// s1[a] = <leaky_relu(Wh[node_a]), a1>;  att = masked softmax over AN=8;
// out_edge[m] = sum_a att[a] * Wh[node_a]     (raw, not relu'd)
// ---------------------------------------------------------------------------
__global__ __launch_bounds__(256) void edge_attention(
    const float* __restrict__ Wh, const float* __restrict__ a1,
    const int* __restrict__ edge_nodes, const int* __restrict__ edge_mask,
    float* __restrict__ out_edge, int M) {
  const int m    = (int)((blockIdx.x * blockDim.x + threadIdx.x) >> 5);
  const int lane = (int)(threadIdx.x & 31);
  if (m >= M) return;

  const float4 a1v = *(const float4*)(a1 + lane * 4);

  float4 ge[AN];
  float  s[AN];
  int    msk[AN];
#pragma unroll
  for (int a = 0; a < AN; ++a) {
    msk[a] = edge_mask[(size_t)m * AN + a];
    const int node = edge_nodes[(size_t)m * AN + a];
    const float4 v = *(const float4*)(Wh + (size_t)node * D + lane * 4);
    ge[a] = v;
    float p = lrelu(v.x) * a1v.x + lrelu(v.y) * a1v.y +
              lrelu(v.z) * a1v.z + lrelu(v.w) * a1v.w;
    p = wave_sum(p);
    s[a] = msk[a] ? p : NEGV;
  }

  float mx = s[0];
#pragma unroll
  for (int a = 1; a < AN; ++a) mx = fmaxf(mx, s[a]);

  float e[AN];
  float den = 0.0f;
#pragma unroll
  for (int a = 0; a < AN; ++a) {
    e[a] = msk[a] ? __expf(s[a] - mx) : 0.0f;
    den += e[a];
  }
  const float inv = 1.0f / den;

  float4 acc = make_float4(0.f, 0.f, 0.f, 0.f);
#pragma unroll
  for (int a = 0; a < AN; ++a) {
    const float w = e[a] * inv;
    acc.x += w * ge[a].x; acc.y += w * ge[a].y;
    acc.z += w * ge[a].z; acc.w += w * ge[a].w;
  }
  *(float4*)(out_edge + (size_t)m * D + lane * 4) = acc;
}

// ---------------------------------------------------------------------------
// Node attention: one wave32 per node. AE=16 incident edges.
// s2[e] = <Wf[edge_e], a2[:D]> + <node_embs[n], a2[D:]>;  masked softmax;
// out_node[n] = sum_e att[e] * Wf[edge_e]
// ---------------------------------------------------------------------------
__global__ __launch_bounds__(256) void node_attention(
    const float* __restrict__ Wf, const float* __restrict__ node_embs,
    const float* __restrict__ a2, const int* __restrict__ node_edges,
    const int* __restrict__ node_mask, float* __restrict__ out_node, int N) {
  const int n    = (int)((blockIdx.x * blockDim.x + threadIdx.x) >> 5);
  const int lane = (int)(threadIdx.x & 31);
  if (n >= N) return;

  const float4 a2f = *(const float4*)(a2 + lane * 4);
  const float4 a2h = *(const float4*)(a2 + D + lane * 4);
  const float4 ne  = *(const float4*)(node_embs + (size_t)n * D + lane * 4);
  const float  h2  = wave_sum(ne.x * a2h.x + ne.y * a2h.y +
                              ne.z * a2h.z + ne.w * a2h.w);

  float4 gn[AE];
  float  s[AE];
  int    msk[AE];
#pragma unroll
  for (int a = 0; a < AE; ++a) {
    msk[a] = node_mask[(size_t)n * AE + a];
    const int ed = node_edges[(size_t)n * AE + a];
    const float4 v = *(const float4*)(Wf + (size_t)ed * D + lane * 4);
    gn[a] = v;
    float p = v.x * a2f.x + v.y * a2f.y + v.z * a2f.z + v.w * a2f.w;
    p = wave_sum(p);
    s[a] = msk[a] ? (p + h2) : NEGV;
  }

  float mx = s[0];
#pragma unroll
  for (int a = 1; a < AE; ++a) mx = fmaxf(mx, s[a]);

  float e[AE];
  float den = 0.0f;
#pragma unroll
  for (int a = 0; a < AE; ++a) {
    e[a] = msk[a] ? __expf(s[a] - mx) : 0.0f;
    den += e[a];
  }
  const float inv = 1.0f / den;

  float4 acc = make_float4(0.f, 0.f, 0.f, 0.f);
#pragma unroll
  for (int a = 0; a < AE; ++a) {
    const float w = e[a] * inv;
    acc.x += w * gn[a].x; acc.y += w * gn[a].y;
    acc.z += w * gn[a].z; acc.w += w * gn[a].w;
  }
  *(float4*)(out_node + (size_t)n * D + lane * 4) = acc;
}

// ---------------------------------------------------------------------------
extern "C" void kernel_launch(void* const* d_in, const int* in_sizes, int n_in,
                              void* d_out, int out_size, void* d_ws, size_t ws_size,
                              hipStream_t stream) {
  const float* node_embs = (const float*)d_in[0];
  // d_in[1] (edge_embs) is unused by the reference computation.
  const float* W1        = (const float*)d_in[2];
  const float* W2        = (const float*)d_in[3];
  const float* a1        = (const float*)d_in[4];
  const float* a2        = (const float*)d_in[5];
  const int*   edge_nodes = (const int*)d_in[6];
  const int*   edge_mask  = (const int*)d_in[7];
  const int*   node_edges = (const int*)d_in[8];
  const int*   node_mask  = (const int*)d_in[9];

  const int N = in_sizes[0] / D;   // 50000
  const int M = in_sizes[1] / D;   // 100000

  float* out_node = (float*)d_out;                  // first output (N x D)
  float* out_edge = out_node + (size_t)N * D;       // second output (M x D)

  float* Wh = (float*)d_ws;                         // N x D
  float* Wf = Wh + (size_t)N * D;                   // M x D

  // 1) Wh = node_embs @ W1   (WMMA f32)
  {
    const int tiles  = (N / 16) * (D / 16);
    const int blocks = (tiles + 7) / 8;             // 8 waves / 256-thread block
    gemm_rowmajor_128<<<blocks, 256, 0, stream>>>(node_embs, W1, Wh, N);
  }
  // 2) edge attention -> out_edge
  {
    const int blocks = (M + 7) / 8;                 // 8 edges (waves) / block
    edge_attention<<<blocks, 256, 0, stream>>>(Wh, a1, edge_nodes, edge_mask,
                                               out_edge, M);
  }
  // 3) Wf = out_edge @ W2   (WMMA f32)
  {
    const int tiles  = (M / 16) * (D / 16);
    const int blocks = (tiles + 7) / 8;
    gemm_rowmajor_128<<<blocks, 256, 0, stream>>>(out_edge, W2, Wf, M);
  }
  // 4) node attention -> out_node
  {
    const int blocks = (N + 7) / 8;
    node_attention<<<blocks, 256, 0, stream>>>(Wf, node_embs, a2, node_edges,
                                               node_mask, out_node, N);
  }
}